// LNN_38199439131314
// MI455X (gfx1250) — compile-verified
//
#include <hip/hip_runtime.h>
#include <hip/hip_bf16.h>
#include <cmath>

typedef __attribute__((ext_vector_type(2))) float v2f;
typedef __attribute__((ext_vector_type(8))) float v8f;

static constexpr int BB = 64;     // batch
static constexpr int SS = 4096;   // sequence length
static constexpr int II = 128;    // input dim
static constexpr int HH = 256;    // hidden dim
static constexpr int OO = 128;    // output dim
static constexpr float DTC = 0.01f;

// LDS row pads chosen so row stride ≡ 4 (mod 64 banks) -> conflict-free
// fragment gathers for the 16x16x4 f32 WMMA layouts.
static constexpr int XPAD  = 132; // x tile rows (16 x 128)
static constexpr int WPAD  = 260; // WhT rows (256 x 256, n-major)
static constexpr int HPAD  = 260; // h rows (16 x 256)

// Low 32 bits of a generic pointer to LDS = hardware LDS byte offset
// (flat->LDS mapping uses addr[31:0] per ISA §10.2).
__device__ __forceinline__ unsigned lds_off(const void* p) {
  return (unsigned)(uintptr_t)p;
}

// ---------------------------------------------------------------------------
// Kernel 1: u[s, b, :] = x[b, s, :] @ W[0:128, :] + bias      (pure GEMM)
//   M = B*S = 262144 rows, K = 128, N = 256.
//   One block = one 16-row M tile; 8 waves x 2 N-tiles each.
//   x tile staged via ASYNCcnt-tracked global_load_async_to_lds_b128.
// ---------------------------------------------------------------------------
__global__ __launch_bounds__(256) void ltc_proj(const float* __restrict__ x,
                                                const float* __restrict__ W,
                                                const float* __restrict__ bias,
                                                float* __restrict__ u) {
  __shared__ float xs[16 * XPAD];
  const int tid  = threadIdx.x;
  const int lane = tid & 31;
  const int wid  = tid >> 5;
  const int mBase = blockIdx.x * 16;

  // Async-stage the 16x128 x tile. Each thread: row = tid>>4, 16B chunk =
  // tid&15, two B128s covering bytes [chunk*16, chunk*16+16) and +256.
  // INST_OFFSET is added to BOTH the LDS and global addresses, so the
  // row-padded LDS destination stays consistent.
  {
    const int row   = tid >> 4;
    const int chunk = tid & 15;
    const unsigned ldsAddr = lds_off(xs) + (unsigned)(row * XPAD * 4 + chunk * 16);
    const unsigned gOff    = (unsigned)((mBase + row) * (II * 4) + chunk * 16);
    asm volatile(
        "global_load_async_to_lds_b128 %0, %1, %2\n\t"
        "global_load_async_to_lds_b128 %0, %1, %2 offset:256"
        :: "v"(ldsAddr), "v"(gOff), "s"(x) : "memory");
  }
  asm volatile("s_wait_asynccnt 0x0" ::: "memory");
  __syncthreads();

  const int halfSel = lane >> 4;   // 0: lanes 0-15, 1: lanes 16-31
  const int ml      = lane & 15;   // A row / B,C,D column within tile

  const int nt0  = wid * 2, nt1 = wid * 2 + 1;
  const int col0 = nt0 * 16 + ml;
  const int col1 = nt1 * 16 + ml;

  v8f acc0 = {};
  v8f acc1 = {};
  for (int kk = 0; kk < II; kk += 4) {
    const int k0 = kk + halfSel * 2;
    v2f a;  a.x  = xs[ml * XPAD + k0];
            a.y  = xs[ml * XPAD + k0 + 1];
    v2f b0; b0.x = W[(size_t)k0 * HH + col0];
            b0.y = W[(size_t)(k0 + 1) * HH + col0];
    v2f b1; b1.x = W[(size_t)k0 * HH + col1];
            b1.y = W[(size_t)(k0 + 1) * HH + col1];
    acc0 = __builtin_amdgcn_wmma_f32_16x16x4_f32(false, a, false, b0,
                                                 (short)0, acc0, false, false);
    acc1 = __builtin_amdgcn_wmma_f32_16x16x4_f32(false, a, false, b1,
                                                 (short)0, acc1, false, false);
  }

  const float bv0 = bias[col0];
  const float bv1 = bias[col1];
  const int bIdx  = mBase / SS;     // x row m = b*S + s
  const int sBase = mBase % SS;     // 16 | S, so whole tile shares bIdx

  for (int r = 0; r < 8; ++r) {
    const int Mrow = r + halfSel * 8;              // C/D layout: VGPR r -> M
    const size_t o = ((size_t)(sBase + Mrow) * BB + bIdx) * HH;
    u[o + col0] = acc0[r] + bv0;
    u[o + col1] = acc1[r] + bv1;
  }
}

// ---------------------------------------------------------------------------
// Kernel 2: sequential LTC scan. 4 blocks x 16 batch rows. 16 waves: one
// 16x16 N-tile of h each. Wh resident (transposed) in LDS for all 4096 steps.
// Per-step u block (16 KB, contiguous in [S,B,H]) double-buffered into LDS
// with async B128 copies overlapping the WMMA chain.
// ---------------------------------------------------------------------------
__global__ __launch_bounds__(512) void ltc_recur(const float* __restrict__ u,
                                                 const float* __restrict__ W,
                                                 const float* __restrict__ tau,
                                                 const float* __restrict__ Avec,
                                                 const float* __restrict__ Wo,
                                                 const float* __restrict__ bo,
                                                 float* __restrict__ out) {
  extern __shared__ float lds[];
  float* WhT  = lds;                        // [256][WPAD], WhT[n][k] = Wh[k][n]
  float* hsm  = WhT + HH * WPAD;            // [16][HPAD]
  float* ubuf = hsm + 16 * HPAD;            // [2][16*256] double buffer

  const int tid  = threadIdx.x;
  const int lane = tid & 31;
  const int wid  = tid >> 5;                // 0..15
  const int bBase = blockIdx.x * 16;
  const int halfSel = lane >> 4;
  const int ml      = lane & 15;

  // Load Wh = W[128:384, :] transposed into LDS (one-time, 260 KB).
  for (int i = tid; i < HH * HH; i += 512) {
    const int n = i >> 8, k = i & 255;
    WhT[n * WPAD + k] = W[(size_t)(II + k) * HH + n];
  }
  // h0 = 0
  for (int i = tid; i < 16 * HPAD; i += 512) hsm[i] = 0.0f;

  // Prime buffer 0 with u[t=0]: 16 KB = 512 threads x 2 B128 chunks.
  {
    const float* g0 = u + (size_t)bBase * HH;       // t = 0
    const unsigned ldsAddr = lds_off(ubuf) + (unsigned)(tid * 16);
    const unsigned gOff    = (unsigned)(tid * 16);
    asm volatile(
        "global_load_async_to_lds_b128 %0, %1, %2\n\t"
        "global_load_async_to_lds_b128 %0, %1, %2 offset:8192"
        :: "v"(ldsAddr), "v"(gOff), "s"(g0) : "memory");
  }
  asm volatile("s_wait_asynccnt 0x0" ::: "memory");
  __syncthreads();

  const int gcol     = wid * 16 + ml;       // this wave's hidden column
  const float aA     = Avec[gcol];
  const float invTau = 1.0f / tau[gcol];

  for (int t = 0; t < SS; ++t) {
    const int p = t & 1;

    // Kick off async fill of the other buffer with u[t+1] (uniform branch).
    if (t + 1 < SS) {
      const float* gn = u + ((size_t)(t + 1) * BB + bBase) * HH;
      const unsigned ldsAddr =
          lds_off(ubuf) + (unsigned)((1 - p) * 16 * HH * 4 + tid * 16);
      const unsigned gOff = (unsigned)(tid * 16);
      asm volatile(
          "global_load_async_to_lds_b128 %0, %1, %2\n\t"
          "global_load_async_to_lds_b128 %0, %1, %2 offset:8192"
          :: "v"(ldsAddr), "v"(gOff), "s"(gn) : "memory");
    }

    // pre = (h @ Wh) tile at cols [wid*16, wid*16+16).
    // Two accumulators (even/odd K pairs) -> 2-way WMMA ILP on the chain.
    v8f accE = {};
    v8f accO = {};
    for (int kk = 0; kk < HH; kk += 8) {
      const int ke = kk + halfSel * 2;
      const int ko = kk + 4 + halfSel * 2;
      v2f ae; ae.x = hsm[ml * HPAD + ke];
              ae.y = hsm[ml * HPAD + ke + 1];
      v2f be; be.x = WhT[gcol * WPAD + ke];
              be.y = WhT[gcol * WPAD + ke + 1];
      v2f ao; ao.x = hsm[ml * HPAD + ko];
              ao.y = hsm[ml * HPAD + ko + 1];
      v2f bov; bov.x = WhT[gcol * WPAD + ko];
               bov.y = WhT[gcol * WPAD + ko + 1];
      accE = __builtin_amdgcn_wmma_f32_16x16x4_f32(false, ae, false, be,
                                                   (short)0, accE, false, false);
      accO = __builtin_amdgcn_wmma_f32_16x16x4_f32(false, ao, false, bov,
                                                   (short)0, accO, false, false);
    }

    const float* ub = ubuf + p * 16 * HH;   // this step's staged u block
    float hn[8];
    for (int r = 0; r < 8; ++r) {
      const int Mrow = r + halfSel * 8;
      const float pre  = accE[r] + accO[r] + ub[Mrow * HH + gcol];
      const float f    = tanhf(pre);
      const float hold = hsm[Mrow * HPAD + gcol];
      // fused semi-implicit LTC step
      hn[r] = (hold + DTC * f * aA) / (1.0f + DTC * (invTau + f));
    }

    __syncthreads();    // all waves done reading h (fragments + update)
    for (int r = 0; r < 8; ++r) {
      const int Mrow = r + halfSel * 8;
      hsm[Mrow * HPAD + gcol] = hn[r];
    }
    // Next step's u buffer must be complete before anyone reads it.
    asm volatile("s_wait_asynccnt 0x0" ::: "memory");
    __syncthreads();
  }

  // out = h_final @ Wo + bo : [16,256] @ [256,128]; 8 N-tiles -> waves 0..7
  // (branch is wave-uniform, so EXEC stays all-ones inside for WMMA)
  if (wid < 8) {
    const int ocol = wid * 16 + ml;
    v8f acc = {};
    for (int kk = 0; kk < HH; kk += 4) {
      const int k0 = kk + halfSel * 2;
      v2f a; a.x = hsm[ml * HPAD + k0];
             a.y = hsm[ml * HPAD + k0 + 1];
      v2f b; b.x = Wo[(size_t)k0 * OO + ocol];
             b.y = Wo[(size_t)(k0 + 1) * OO + ocol];
      acc = __builtin_amdgcn_wmma_f32_16x16x4_f32(false, a, false, b,
                                                  (short)0, acc, false, false);
    }
    const float bv = bo[ocol];
    for (int r = 0; r < 8; ++r) {
      const int Mrow = r + halfSel * 8;
      out[(size_t)(bBase + Mrow) * OO + ocol] = acc[r] + bv;
    }
  }
}

// ---------------------------------------------------------------------------
extern "C" void kernel_launch(void* const* d_in, const int* in_sizes, int n_in,
                              void* d_out, int out_size, void* d_ws, size_t ws_size,
                              hipStream_t stream) {
  const float* x    = (const float*)d_in[0];   // [B,S,I]
  const float* W    = (const float*)d_in[1];   // [I+H,H]
  const float* bias = (const float*)d_in[2];   // [H]
  const float* tau  = (const float*)d_in[3];   // [H]
  const float* A    = (const float*)d_in[4];   // [H]
  const float* Wo   = (const float*)d_in[5];   // [H,O]
  const float* bo   = (const float*)d_in[6];   // [O]
  float* out = (float*)d_out;                  // [B,O]
  float* u   = (float*)d_ws;                   // [S,B,H] pre-activations

  (void)in_sizes; (void)n_in; (void)out_size; (void)ws_size;

  // Phase 1: parallel input projection (no sequential dependency).
  ltc_proj<<<(BB * SS) / 16, 256, 0, stream>>>(x, W, bias, u);

  // Phase 2: batch-parallel sequential scan; Wh + h + u double-buffer in LDS.
  const size_t ldsBytes =
      (size_t)(HH * WPAD + 16 * HPAD + 2 * 16 * HH) * sizeof(float); // ~308 KB
  hipFuncSetAttribute((const void*)ltc_recur,
                      hipFuncAttributeMaxDynamicSharedMemorySize,
                      (int)ldsBytes);
  ltc_recur<<<BB / 16, 512, ldsBytes, stream>>>(u, W, tau, A, Wo, bo, out);
}